// SGLCN_85718957293636
// MI455X (gfx1250) — compile-verified
//
#include <hip/hip_runtime.h>
#include <hip/hip_bf16.h>

typedef __attribute__((ext_vector_type(2))) float v2f;
typedef __attribute__((ext_vector_type(8))) float v8f;

#define N_NODES 50000
#define N_EDGES 1600000
#define D_IN    256
#define H_GL    32
#define H_GCN   64
#define D_OUT_  16

// ---------------------------------------------------------------------------
// Kernel 1: fused dense GEMMs  h = x @ Wgl  (N x 32)  and  xw1 = x @ W1 (N x 64)
// One wave per 16-row tile; 6 accumulator tiles; V_WMMA_F32_16X16X4_F32.
// A layout (16x4 f32): lane m=lane&15 holds row m; vgpr pair = {K=kb, K=kb+1},
// kb = k + 2*(lane>>4).  B layout (4x16): lane n=lane&15 holds col n, same K split.
// C/D layout: acc[i] -> row = i + 8*(lane>>4), col = lane&15.
// ---------------------------------------------------------------------------
__global__ void sglcn_gemm_fused(const float* __restrict__ x,
                                 const float* __restrict__ Wgl,
                                 const float* __restrict__ W1,
                                 float* __restrict__ h,
                                 float* __restrict__ xw1) {
  const int lane = threadIdx.x & 31;
  const int wave = threadIdx.x >> 5;
  const int tile = blockIdx.x * 8 + wave;
  if (tile * 16 >= N_NODES) return;            // wave-uniform: EXEC stays full
  const int row0 = tile * 16;
  const int mcol = lane & 15;
  const int half = lane >> 4;

  const float* xrow = x + (size_t)(row0 + mcol) * D_IN;

  v8f acc0 = {}; v8f acc1 = {};                         // h cols 0-15, 16-31
  v8f acc2 = {}; v8f acc3 = {}; v8f acc4 = {}; v8f acc5 = {}; // xw1 cols 0-63

  for (int k = 0; k < D_IN; k += 4) {
    const int kb = k + 2 * half;
    v2f a;
    a.x = xrow[kb];
    a.y = xrow[kb + 1];

    v2f b;
    b.x = Wgl[kb * H_GL + mcol];
    b.y = Wgl[(kb + 1) * H_GL + mcol];
    acc0 = __builtin_amdgcn_wmma_f32_16x16x4_f32(false, a, false, b, (short)0, acc0, false, false);

    b.x = Wgl[kb * H_GL + 16 + mcol];
    b.y = Wgl[(kb + 1) * H_GL + 16 + mcol];
    acc1 = __builtin_amdgcn_wmma_f32_16x16x4_f32(false, a, false, b, (short)0, acc1, false, false);

    b.x = W1[kb * H_GCN + mcol];
    b.y = W1[(kb + 1) * H_GCN + mcol];
    acc2 = __builtin_amdgcn_wmma_f32_16x16x4_f32(false, a, false, b, (short)0, acc2, false, false);

    b.x = W1[kb * H_GCN + 16 + mcol];
    b.y = W1[(kb + 1) * H_GCN + 16 + mcol];
    acc3 = __builtin_amdgcn_wmma_f32_16x16x4_f32(false, a, false, b, (short)0, acc3, false, false);

    b.x = W1[kb * H_GCN + 32 + mcol];
    b.y = W1[(kb + 1) * H_GCN + 32 + mcol];
    acc4 = __builtin_amdgcn_wmma_f32_16x16x4_f32(false, a, false, b, (short)0, acc4, false, false);

    b.x = W1[kb * H_GCN + 48 + mcol];
    b.y = W1[(kb + 1) * H_GCN + 48 + mcol];
    acc5 = __builtin_amdgcn_wmma_f32_16x16x4_f32(false, a, false, b, (short)0, acc5, false, false);
  }

  const int rs = 8 * half;
#pragma unroll
  for (int i = 0; i < 8; ++i) {
    const int r = row0 + i + rs;
    h[(size_t)r * H_GL + mcol]        = acc0[i];
    h[(size_t)r * H_GL + 16 + mcol]   = acc1[i];
    xw1[(size_t)r * H_GCN + mcol]      = acc2[i];
    xw1[(size_t)r * H_GCN + 16 + mcol] = acc3[i];
    xw1[(size_t)r * H_GCN + 32 + mcol] = acc4[i];
    xw1[(size_t)r * H_GCN + 48 + mcol] = acc5[i];
  }
}

// ---------------------------------------------------------------------------
// Kernel 2: edge scoring. One wave per edge; lane = feature dim (H_GL == 32).
// score = relu( sum_k |h[r][k]-h[c][k]| * a[k] ); row-max via uint atomicMax
// (valid: all scores >= 0 after relu; nonneg floats order as uint bit patterns).
// ---------------------------------------------------------------------------
__global__ void sglcn_edge_score(const float* __restrict__ h,
                                 const int* __restrict__ rows,
                                 const int* __restrict__ cols,
                                 const float* __restrict__ avec,
                                 float* __restrict__ score,
                                 float* __restrict__ rowmax) {
  const int e = (int)((blockIdx.x * (unsigned)blockDim.x + threadIdx.x) >> 5);
  const int lane = threadIdx.x & 31;
  if (e >= N_EDGES) return;
  const int r = rows[e];
  const int c = cols[e];
  float v = fabsf(h[(size_t)r * H_GL + lane] - h[(size_t)c * H_GL + lane]) * avec[lane];
#pragma unroll
  for (int off = 16; off > 0; off >>= 1) v += __shfl_xor(v, off, 32);
  if (lane == 0) {
    const float sv = fmaxf(v, 0.0f);
    score[e] = sv;
    atomicMax((unsigned int*)(rowmax + r), __float_as_uint(sv));
  }
}

// ---------------------------------------------------------------------------
// Kernel 3: softmax numerator: adj[e] = exp(score - max[row]); rowsum += adj[e]
// ---------------------------------------------------------------------------
__global__ void sglcn_soft_exp(const int* __restrict__ rows,
                               const float* __restrict__ rowmax,
                               float* __restrict__ adj,
                               float* __restrict__ rowsum) {
  const int e = blockIdx.x * blockDim.x + threadIdx.x;
  if (e >= N_EDGES) return;
  const int r = rows[e];
  const float ev = __expf(adj[e] - rowmax[r]);
  adj[e] = ev;
  atomicAdd(rowsum + r, ev);
}

// Kernel 4: normalize: adj[e] /= rowsum[row]
__global__ void sglcn_soft_div(const int* __restrict__ rows,
                               const float* __restrict__ rowsum,
                               float* __restrict__ adj) {
  const int e = blockIdx.x * blockDim.x + threadIdx.x;
  if (e >= N_EDGES) return;
  adj[e] /= rowsum[rows[e]];
}

// ---------------------------------------------------------------------------
// Kernel 5: SpMM1: x1[r][:] += adj[e] * xw1[c][:]   (64 dims; wave/edge,
// float2 per lane, 2 atomic f32 adds -> GLOBAL_ATOMIC_ADD_F32)
// ---------------------------------------------------------------------------
__global__ void sglcn_spmm1(const int* __restrict__ rows,
                            const int* __restrict__ cols,
                            const float* __restrict__ adj,
                            const float* __restrict__ xw1,
                            float* __restrict__ x1) {
  const int e = (int)((blockIdx.x * (unsigned)blockDim.x + threadIdx.x) >> 5);
  const int lane = threadIdx.x & 31;
  if (e >= N_EDGES) return;
  const int r = rows[e];
  const int c = cols[e];
  const float val = adj[e];
  const float2 w = ((const float2*)(xw1 + (size_t)c * H_GCN))[lane];
  float* dst = x1 + (size_t)r * H_GCN + 2 * lane;
  atomicAdd(dst,     val * w.x);
  atomicAdd(dst + 1, val * w.y);
}

// ---------------------------------------------------------------------------
// Kernel 6: GEMM2: xw2 = relu(x1) @ W2   (N x 64 @ 64 x 16), one WMMA tile/wave
// ---------------------------------------------------------------------------
__global__ void sglcn_gemm2(const float* __restrict__ x1,
                            const float* __restrict__ W2,
                            float* __restrict__ xw2) {
  const int lane = threadIdx.x & 31;
  const int wave = threadIdx.x >> 5;
  const int tile = blockIdx.x * 8 + wave;
  if (tile * 16 >= N_NODES) return;
  const int row0 = tile * 16;
  const int mcol = lane & 15;
  const int half = lane >> 4;
  const float* xrow = x1 + (size_t)(row0 + mcol) * H_GCN;

  v8f acc = {};
  for (int k = 0; k < H_GCN; k += 4) {
    const int kb = k + 2 * half;
    v2f a;
    a.x = fmaxf(xrow[kb], 0.0f);          // relu fused into A load
    a.y = fmaxf(xrow[kb + 1], 0.0f);
    v2f b;
    b.x = W2[kb * D_OUT_ + mcol];
    b.y = W2[(kb + 1) * D_OUT_ + mcol];
    acc = __builtin_amdgcn_wmma_f32_16x16x4_f32(false, a, false, b, (short)0, acc, false, false);
  }
  const int rs = 8 * half;
#pragma unroll
  for (int i = 0; i < 8; ++i) {
    xw2[(size_t)(row0 + i + rs) * D_OUT_ + mcol] = acc[i];
  }
}

// ---------------------------------------------------------------------------
// Kernel 7: SpMM2: out[r][:] += adj[e] * xw2[c][:]   (16 dims; 16 lanes/edge)
// ---------------------------------------------------------------------------
__global__ void sglcn_spmm2(const int* __restrict__ rows,
                            const int* __restrict__ cols,
                            const float* __restrict__ adj,
                            const float* __restrict__ xw2,
                            float* __restrict__ out) {
  const unsigned gid = blockIdx.x * (unsigned)blockDim.x + threadIdx.x;
  const int e = (int)(gid >> 4);
  const int d = gid & 15;
  if (e >= N_EDGES) return;
  const int r = rows[e];
  const int c = cols[e];
  atomicAdd(out + (size_t)r * D_OUT_ + d, adj[e] * xw2[(size_t)c * D_OUT_ + d]);
}

// ---------------------------------------------------------------------------
extern "C" void kernel_launch(void* const* d_in, const int* in_sizes, int n_in,
                              void* d_out, int out_size, void* d_ws, size_t ws_size,
                              hipStream_t stream) {
  const float* x   = (const float*)d_in[0];
  const int*   edg = (const int*)  d_in[1];
  const float* Wgl = (const float*)d_in[2];
  const float* av  = (const float*)d_in[3];
  const float* W1  = (const float*)d_in[4];
  const float* W2  = (const float*)d_in[5];
  const int* rows = edg;
  const int* cols = edg + N_EDGES;

  // Output tuple layout: output [N,16] | adj_vals [E] | h [N,32]
  float* out = (float*)d_out;
  float* adj = out + (size_t)N_NODES * D_OUT_;
  float* h   = adj + (size_t)N_EDGES;

  // Workspace layout (floats): xw1 | x1 | xw2 | rowmax | rowsum
  float* ws     = (float*)d_ws;
  float* xw1    = ws;
  float* x1     = xw1    + (size_t)N_NODES * H_GCN;
  float* xw2    = x1     + (size_t)N_NODES * H_GCN;
  float* rowmax = xw2    + (size_t)N_NODES * D_OUT_;
  float* rowsum = rowmax + (size_t)N_NODES;

  // Zero accumulators every call (atomics accumulate; graph replays don't re-poison)
  hipMemsetAsync(out,    0, sizeof(float) * (size_t)N_NODES * D_OUT_, stream);
  hipMemsetAsync(x1,     0, sizeof(float) * (size_t)N_NODES * H_GCN,  stream);
  hipMemsetAsync(rowmax, 0, sizeof(float) * (size_t)N_NODES * 2,      stream); // rowmax + rowsum

  const int TILES = (N_NODES + 15) / 16;           // 3125, exact
  const int GEMM_BLOCKS = (TILES + 7) / 8;         // 391

  sglcn_gemm_fused<<<GEMM_BLOCKS, 256, 0, stream>>>(x, Wgl, W1, h, xw1);

  const int WPE_BLOCKS = (int)(((size_t)N_EDGES * 32 + 255) / 256);  // wave per edge
  sglcn_edge_score<<<WPE_BLOCKS, 256, 0, stream>>>(h, rows, cols, av, adj, rowmax);

  const int TPE_BLOCKS = (N_EDGES + 255) / 256;                       // thread per edge
  sglcn_soft_exp<<<TPE_BLOCKS, 256, 0, stream>>>(rows, rowmax, adj, rowsum);
  sglcn_soft_div<<<TPE_BLOCKS, 256, 0, stream>>>(rows, rowsum, adj);

  sglcn_spmm1<<<WPE_BLOCKS, 256, 0, stream>>>(rows, cols, adj, xw1, x1);

  sglcn_gemm2<<<GEMM_BLOCKS, 256, 0, stream>>>(x1, W2, xw2);

  const int HPE_BLOCKS = (int)(((size_t)N_EDGES * 16 + 255) / 256);   // 16 lanes per edge
  sglcn_spmm2<<<HPE_BLOCKS, 256, 0, stream>>>(rows, cols, adj, xw2, out);
}